// DiodeClipper_4990751998187
// MI455X (gfx1250) — compile-verified
//
#include <hip/hip_runtime.h>
#include <math.h>

// Diode clipper constants (match the JAX reference, computed in double, cast to f32)
// SR=44100, R=1000, C=3.3e-8, I_S=2.52e-9, V_T=0.02583
// OMEGA = 1/(R*C), K = 1/SR, SINH_C = 2*I_S/C
// A        = 0.5*K*OMEGA
// halfBc   = 0.25*K*SINH_C              (coefficient on (e - 1/e) = 2 sinh)
// halfBoVT = 0.25*K*SINH_C / V_T        (coefficient on (e + 1/e) = 2 cosh)
// twoA     = K*OMEGA
namespace dc {
constexpr double SR     = 44100.0;
constexpr double R      = 1000.0;
constexpr double C      = 3.3e-08;
constexpr double I_S    = 2.52e-09;
constexpr double V_T    = 0.02583;
constexpr double K      = 1.0 / SR;
constexpr double OMEGA  = 1.0 / (R * C);
constexpr double SINH_C = 2.0 * I_S / C;

constexpr float A_F        = (float)(0.5 * K * OMEGA);           // 0.34357...
constexpr float HALFBC_F   = (float)(0.25 * K * SINH_C);         // 8.658e-7
constexpr float HALFBOVT_F = (float)(0.25 * K * SINH_C / V_T);   // 3.352e-5
constexpr float TWOA_F     = (float)(K * OMEGA);                 // 0.68714...
constexpr float INV_VT_F   = (float)(1.0 / V_T);                 // 38.714...
constexpr float ONE_PLUS_A = (float)(1.0 + 0.5 * K * OMEGA);     // 1.34357...
constexpr float A_MINUS_1  = (float)(0.5 * K * OMEGA - 1.0);     // -0.65642...
constexpr float TOL_F      = 1e-9f;
constexpr int   MAX_ITER   = 50;
} // namespace dc

// One lane per batch element. B=128 -> 4 wave32 waves, one per block so the
// scheduler spreads the four serial dependency chains across WGPs/SIMDs.
__global__ __launch_bounds__(32) void
diode_clipper_scan(const float* __restrict__ x,    // [B, T]
                   const float* __restrict__ h0,   // [B]
                   float* __restrict__ states,     // [B, T]
                   float* __restrict__ h_final,    // [B]
                   int B, int T)
{
    const int b = blockIdx.x * 32 + threadIdx.x;
    if (b >= B) return;

    const float4* __restrict__ xin  = reinterpret_cast<const float4*>(x + (size_t)b * T);
    float4* __restrict__       outp = reinterpret_cast<float4*>(states + (size_t)b * T);

    float h = h0[b];

    const int nChunks = T >> 2;   // T is a multiple of 4 (16384)

    for (int c = 0; c < nChunks; ++c) {
        // Stream the input ahead of the serial compute (global_prefetch_b8).
        if (c + 16 < nChunks) {
            __builtin_prefetch((const void*)&xin[c + 16], 0, 0);
        }

        const float4 xv = xin[c];
        float4 ov;
        float xs[4] = { xv.x, xv.y, xv.z, xv.w };
        float os[4];

        #pragma unroll
        for (int j = 0; j < 4; ++j) {
            const float xt = xs[j];

            // exp(+-h/V_T): used for p AND reused exactly by Newton iter 0
            // (v0 == h_prev, so these are the same values the reference's
            //  first iteration recomputes -- fold them, saving 2 trans ops
            //  on the serial critical path every time step).
            float w  = h * dc::INV_VT_F;
            float e  = __expf(w);        // v_exp_f32 (trans pipe)
            float em = __expf(-w);       // independent -> issues back-to-back

            // p = 0.5*K*f_nl(h) - K*OMEGA*x - h
            //   = (A-1)*h + halfBc*(e - 1/e) - twoA*x
            const float p = fmaf(dc::A_MINUS_1, h,
                                 fmaf(dc::HALFBC_F, e - em, -dc::TWOA_F * xt));

            // Newton-Raphson on g(v) = (1+A)v + halfBc*(e - 1/e) + p, v0 = h.
            // Loop invariant: (e, em) == exp(+-v/V_T) at loop top.
            float v = h;
            #pragma unroll 1
            for (int i = 0; i < dc::MAX_ITER; ++i) {
                const float g  = fmaf(dc::ONE_PLUS_A, v,
                                      fmaf(dc::HALFBC_F, e - em, p));
                const float gd = fmaf(dc::HALFBOVT_F, e + em, dc::ONE_PLUS_A);
                // Newton is self-correcting: 1-ulp v_rcp_f32 instead of IEEE divide
                const float s = g * __builtin_amdgcn_rcpf(gd);
                v -= s;
                if (fabsf(s) <= dc::TOL_F) break;   // EXEC-masked per-lane exit
                // Only recompute the transcendentals when we must iterate again.
                w  = v * dc::INV_VT_F;
                e  = __expf(w);
                em = __expf(-w);
            }
            h = v;
            os[j] = v;
        }

        ov.x = os[0]; ov.y = os[1]; ov.z = os[2]; ov.w = os[3];
        outp[c] = ov;   // global_store_b128 every 4 steps
    }

    h_final[b] = h;
}

extern "C" void kernel_launch(void* const* d_in, const int* in_sizes, int n_in,
                              void* d_out, int out_size, void* d_ws, size_t ws_size,
                              hipStream_t stream)
{
    (void)n_in; (void)d_ws; (void)ws_size; (void)out_size;

    const float* x  = (const float*)d_in[0];   // [B, T, 1] float32
    const float* h0 = (const float*)d_in[1];   // [B, 1]    float32

    const int B = in_sizes[1];                 // 128
    const int T = in_sizes[0] / B;             // 16384

    float* states  = (float*)d_out;            // first output: [B, T, 1]
    float* h_final = states + (size_t)B * T;   // second output: [B, 1]

    const int threads = 32;                    // one wave32 per block
    const int blocks  = (B + threads - 1) / threads;   // 4 blocks
    diode_clipper_scan<<<blocks, threads, 0, stream>>>(x, h0, states, h_final, B, T);
}